// TemporalPositionWiseMaxPooling_83313775607962
// MI455X (gfx1250) — compile-verified
//
#include <hip/hip_runtime.h>

// Problem constants (reference: B=16, T=1024, C=1024, fp32 in/out).
#define B_   16
#define T_   1024
#define C_   1024
#define NCH  16          // time chunks
#define CH   64          // T_/NCH steps per chunk
#define BLK  256         // threads per block (8 wave32s)

// ---------------------------------------------------------------------------
// CDNA5 async data mover: per-lane 16B global -> LDS copy, tracked by ASYNCcnt.
// Low 32 bits of a generic pointer to LDS are the workgroup-relative DS byte
// address (LDS aperture: addr[31:0] == LDS offset), which is what the VDST
// operand of global_load_async_to_lds_* expects.
// ---------------------------------------------------------------------------
__device__ __forceinline__ void async_load_b128_to_lds(const void* g, void* lds) {
  unsigned lds_off = (unsigned)(unsigned long long)lds;
  asm volatile("global_load_async_to_lds_b128 %0, %1, off"
               :: "v"(lds_off), "v"(g)
               : "memory");
}

__device__ __forceinline__ void wait_async_zero() {
#if defined(__has_builtin)
#if __has_builtin(__builtin_amdgcn_s_wait_asynccnt)
  __builtin_amdgcn_s_wait_asynccnt(0);
#else
  asm volatile("s_wait_asynccnt 0" ::: "memory");
#endif
#else
  asm volatile("s_wait_asynccnt 0" ::: "memory");
#endif
}

// ---------------------------------------------------------------------------
// Kernel 1: chunkX[ch][b][c] = sum_{t in chunk ch} x[b][t][c]
// 256K threads, fully coalesced strided reads (adjacent threads -> adjacent c).
// ---------------------------------------------------------------------------
__global__ __launch_bounds__(BLK) void k_chunk_sums(const float* __restrict__ x,
                                                    float* __restrict__ chunkX) {
  const int idx = blockIdx.x * BLK + threadIdx.x;   // flat (ch, b, c)
  const int c  = idx & (C_ - 1);
  const int b  = (idx >> 10) & (B_ - 1);
  const int ch = idx >> 14;
  const float* p = x + ((size_t)b * T_ + (size_t)ch * CH) * C_ + c;
  float s = 0.0f;
#pragma unroll 8
  for (int j = 0; j < CH; ++j) s += p[(size_t)j * C_];
  chunkX[idx] = s;   // layout matches [ch][b][c]
}

// ---------------------------------------------------------------------------
// Kernel 2: chunkA[ch][b][c] = sum_{t in chunk} t * s_t,
// with s_t = (total - prefix_{t-1}) / (T - t).
// x tile (64 x 256 fp32 = 64KB) staged via async DMA into LDS.
// ---------------------------------------------------------------------------
__global__ __launch_bounds__(BLK) void k_chunk_A(const float* __restrict__ x,
                                                 const float* __restrict__ chunkX,
                                                 float* __restrict__ chunkA) {
  __shared__ __align__(16) float tile[CH][BLK];
  __shared__ float invT[CH];

  const int tid   = threadIdx.x;
  const int ch    = blockIdx.y;
  const int t0    = ch * CH;
  const int b     = blockIdx.x >> 2;
  const int cbase = (blockIdx.x & 3) * BLK;
  const int c     = cbase + tid;

  if (tid < CH) invT[tid] = 1.0f / (float)(T_ - (t0 + tid));

  // Stage 64 rows x 1KB: 256 lanes x 16B cover 4 rows per async issue.
  const float* gbase = x + ((size_t)b * T_ + t0) * C_ + cbase;
  const int srow = tid >> 6;          // 0..3
  const int scol = (tid & 63) << 2;   // 0..252, 16B aligned
#pragma unroll
  for (int k = 0; k < 16; ++k) {
    const int row = k * 4 + srow;
    async_load_b128_to_lds(gbase + (size_t)row * C_ + scol, &tile[row][scol]);
  }

  // Exclusive prefix over chunk sums + grand total (fixed order, L2-hot).
  float P = 0.0f, total = 0.0f;
#pragma unroll
  for (int i = 0; i < NCH; ++i) {
    const float v = chunkX[(size_t)i * (B_ * C_) + (size_t)b * C_ + c];
    if (i < ch) P += v;
    total += v;
  }

  wait_async_zero();
  __syncthreads();

  float A = 0.0f;
#pragma unroll 4
  for (int j = 0; j < CH; ++j) {
    const float s = (total - P) * invT[j];
    A = fmaf((float)(t0 + j), s, A);
    P += tile[j][tid];
  }
  chunkA[(size_t)ch * (B_ * C_) + (size_t)b * C_ + c] = A;
}

// ---------------------------------------------------------------------------
// Kernel 3: outputs.  r_0 = s_0;  r_1 = s_1 - s_0;
//           r_t = s_t - a_{t-1}/((t-1)t),  a_t = a_{t-1} + t*s_t.
// a_{chunk_start-1} comes from summing chunkA of earlier chunks (fixed order).
// ---------------------------------------------------------------------------
__global__ __launch_bounds__(BLK) void k_finalize(const float* __restrict__ x,
                                                  const float* __restrict__ chunkX,
                                                  const float* __restrict__ chunkA,
                                                  float* __restrict__ out) {
  __shared__ __align__(16) float tile[CH][BLK];
  __shared__ float invT[CH];
  __shared__ float invD[CH];

  const int tid   = threadIdx.x;
  const int ch    = blockIdx.y;
  const int t0    = ch * CH;
  const int b     = blockIdx.x >> 2;
  const int cbase = (blockIdx.x & 3) * BLK;
  const int c     = cbase + tid;

  if (tid < CH) {
    const int t = t0 + tid;
    invT[tid] = 1.0f / (float)(T_ - t);
    invD[tid] = (t >= 2) ? (1.0f / ((float)(t - 1) * (float)t)) : 0.0f;
  }

  const float* gbase = x + ((size_t)b * T_ + t0) * C_ + cbase;
  const int srow = tid >> 6;
  const int scol = (tid & 63) << 2;
#pragma unroll
  for (int k = 0; k < 16; ++k) {
    const int row = k * 4 + srow;
    async_load_b128_to_lds(gbase + (size_t)row * C_ + scol, &tile[row][scol]);
  }

  float P = 0.0f, total = 0.0f;
#pragma unroll
  for (int i = 0; i < NCH; ++i) {
    const float v = chunkX[(size_t)i * (B_ * C_) + (size_t)b * C_ + c];
    if (i < ch) P += v;
    total += v;
  }
  float A = 0.0f;                    // a_{t0-1}
  for (int i = 0; i < ch; ++i)
    A += chunkA[(size_t)i * (B_ * C_) + (size_t)b * C_ + c];

  wait_async_zero();
  __syncthreads();

  float* obase = out + ((size_t)b * T_ + t0) * C_ + c;
  float s0 = 0.0f;                   // only chunk 0 uses this
#pragma unroll 4
  for (int j = 0; j < CH; ++j) {
    const int t = t0 + j;
    const float s = (total - P) * invT[j];
    float r;
    if (t >= 2)        r = s - A * invD[j];
    else if (t == 1)   r = s - s0;            // r_1 = s_1 - m_0 = s_1 - s_0
    else             { r = s; s0 = s; }       // r_0 = s_0
    obase[(size_t)j * C_] = r;
    A = fmaf((float)t, s, A);                 // a_t = a_{t-1} + t*s_t
    P += tile[j][tid];
  }
}

// ---------------------------------------------------------------------------
extern "C" void kernel_launch(void* const* d_in, const int* in_sizes, int n_in,
                              void* d_out, int out_size, void* d_ws, size_t ws_size,
                              hipStream_t stream) {
  (void)in_sizes; (void)n_in; (void)out_size; (void)ws_size;
  const float* x = (const float*)d_in[0];
  float* out = (float*)d_out;
  float* chunkX = (float*)d_ws;                           // 1 MiB
  float* chunkA = chunkX + (size_t)NCH * B_ * C_;         // +1 MiB (ws >= 2 MiB)

  const dim3 blk(BLK);
  k_chunk_sums<<<dim3((NCH * B_ * C_) / BLK), blk, 0, stream>>>(x, chunkX);

  const dim3 grid2((B_ * C_) / BLK, NCH);                 // (b,c-tile) x chunk
  k_chunk_A  <<<grid2, blk, 0, stream>>>(x, chunkX, chunkA);
  k_finalize <<<grid2, blk, 0, stream>>>(x, chunkX, chunkA, out);
}